// EntropyBottleneckLattice_59785944760412
// MI455X (gfx1250) — compile-verified
//
#include <hip/hip_runtime.h>

// Problem constants (match the reference).
#define CCH 256   // channels C
#define BBB 512   // batch B
#define NNN 128   // noise samples N
#define REC 64    // floats per per-channel parameter record in d_ws (64B-pad)

// Record layout (floats, within rec + c*REC):
//   0: W0[3]   softplus(m0)         (layer 1->3)
//   3: W1[9]   softplus(m1) [i*3+j] (3->3)
//  12: W2[9]   softplus(m2)
//  21: W3[9]   softplus(m3)
//  30: W4[3]   softplus(m4)         (3->1)
//  33: B0[3]  36: B1[3]  39: B2[3]  42: B3[3]  45: B4
//  46: T0[3]  49: T1[3]  52: T2[3]  55: T3[3]   (tanh(f_i))

#define LOG2E 1.4426950408889634f
#define LN2   0.6931471805599453f

__device__ __forceinline__ float fast_exp2(float x) {
#if __has_builtin(__builtin_amdgcn_exp2f)
  return __builtin_amdgcn_exp2f(x);
#else
  return exp2f(x);
#endif
}

__device__ __forceinline__ float fast_log2(float x) {
#if __has_builtin(__builtin_amdgcn_logf)
  return __builtin_amdgcn_logf(x);
#else
  return log2f(x);
#endif
}

__device__ __forceinline__ float fast_rcp(float x) {
#if __has_builtin(__builtin_amdgcn_rcpf)
  return __builtin_amdgcn_rcpf(x);
#else
  return 1.0f / x;
#endif
}

// v_tanh_f32 is NEW on gfx1250 (no hardware tanh on CDNA4) — this kernel's
// throughput limiter is ~200M tanh evaluations, so force the 1-op TRANS path.
__device__ __forceinline__ float fast_tanh(float x) {
#if __has_builtin(__builtin_amdgcn_tanhf)
  return __builtin_amdgcn_tanhf(x);
#else
  // Inline-asm fallback. The trailing v_nop satisfies the ISA §7.4 rule that
  // a TRANS result needs 1 independent op before use (the compiler can't
  // schedule around opaque asm, so we pay it inside).
  float r;
  asm volatile("v_tanh_f32 %0, %1\n\tv_nop" : "=v"(r) : "v"(x));
  return r;
#endif
}

__device__ __forceinline__ float softplus_f(float x) {
  // ln(1 + e^x) = ln2 * log2(1 + 2^(x*log2e)); overflow-safe:
  // large x -> x ; very negative x -> ~e^x.
  return LN2 * fast_log2(1.0f + fast_exp2(x * LOG2E));
}

// Gated-tanh nonlinearity + its derivative factor.
// In:  h = u (pre-activation), d = du/dx, tf = tanh(f)
// Out: h = u + tf*tanh(u), d = d * (1 + tf*(1 - tanh(u)^2))
__device__ __forceinline__ void gact(float& h, float& d, float tf) {
  const float t = fast_tanh(h);
  d = d * fmaf(tf, fmaf(-t, t, 1.0f), 1.0f);
  h = fmaf(tf, t, h);
}

// ---------------------------------------------------------------------------
// Main kernel (defined FIRST so it leads the disassembly): Monte-Carlo
// density evaluation. One block == one channel slice; channel c is UNIFORM
// per block, so the 58 parameter loads and the per-iteration noise load are
// wave-uniform -> SMEM s_load, and the tower FMAs take SGPR weight operands
// (tiny VGPR footprint -> max occupancy). Each thread owns one (b, c) output
// and loops over the 128 noise samples, propagating the value and the
// analytic d/dx through the 1->3->3->3->3->1 tower entirely in registers.
// ---------------------------------------------------------------------------
#define BLK 128
#define BLOCKS_PER_C (BBB / BLK)   // 4

__global__ void __launch_bounds__(BLK)
ebl_main(const float* __restrict__ inputs,   // (B, C)
         const float* __restrict__ noise,    // (N, C)
         const float* __restrict__ rec,      // (C, REC)
         float* __restrict__ out) {          // (B, C)
  const int c = (int)blockIdx.x / BLOCKS_PER_C;                 // uniform
  const int b = ((int)blockIdx.x % BLOCKS_PER_C) * BLK + (int)threadIdx.x;

  const float* __restrict__ r = rec + c * REC;                  // uniform addr
  float W0[3], W1[9], W2[9], W3[9], W4[3];
  float B0[3], B1[3], B2[3], B3[3], B4;
  float T0[3], T1[3], T2[3], T3[3];
#pragma unroll
  for (int i = 0; i < 3; ++i) W0[i] = r[0 + i];
#pragma unroll
  for (int i = 0; i < 9; ++i) W1[i] = r[3 + i];
#pragma unroll
  for (int i = 0; i < 9; ++i) W2[i] = r[12 + i];
#pragma unroll
  for (int i = 0; i < 9; ++i) W3[i] = r[21 + i];
#pragma unroll
  for (int i = 0; i < 3; ++i) W4[i] = r[30 + i];
#pragma unroll
  for (int i = 0; i < 3; ++i) B0[i] = r[33 + i];
#pragma unroll
  for (int i = 0; i < 3; ++i) B1[i] = r[36 + i];
#pragma unroll
  for (int i = 0; i < 3; ++i) B2[i] = r[39 + i];
#pragma unroll
  for (int i = 0; i < 3; ++i) B3[i] = r[42 + i];
  B4 = r[45];
#pragma unroll
  for (int i = 0; i < 3; ++i) T0[i] = r[46 + i];
#pragma unroll
  for (int i = 0; i < 3; ++i) T1[i] = r[49 + i];
#pragma unroll
  for (int i = 0; i < 3; ++i) T2[i] = r[52 + i];
#pragma unroll
  for (int i = 0; i < 3; ++i) T3[i] = r[55 + i];

  const float x0 = inputs[b * CCH + c];
  const float* __restrict__ np = noise + c;   // uniform: np[n*CCH] per wave

  float acc = 0.0f;
#pragma unroll 2
  for (int n = 0; n < NNN; ++n) {
    const float x = x0 + np[n * CCH];

    // ----- layer 0: 1 -> 3, then gated tanh -----
    float h0 = fmaf(W0[0], x, B0[0]);
    float h1 = fmaf(W0[1], x, B0[1]);
    float h2 = fmaf(W0[2], x, B0[2]);
    float d0 = W0[0], d1 = W0[1], d2 = W0[2];
    gact(h0, d0, T0[0]); gact(h1, d1, T0[1]); gact(h2, d2, T0[2]);

    // ----- layer 1: 3 -> 3 -----
    {
      float u0 = fmaf(W1[0], h0, fmaf(W1[1], h1, fmaf(W1[2], h2, B1[0])));
      float u1 = fmaf(W1[3], h0, fmaf(W1[4], h1, fmaf(W1[5], h2, B1[1])));
      float u2 = fmaf(W1[6], h0, fmaf(W1[7], h1, fmaf(W1[8], h2, B1[2])));
      float e0 = fmaf(W1[0], d0, fmaf(W1[1], d1, W1[2] * d2));
      float e1 = fmaf(W1[3], d0, fmaf(W1[4], d1, W1[5] * d2));
      float e2 = fmaf(W1[6], d0, fmaf(W1[7], d1, W1[8] * d2));
      h0 = u0; h1 = u1; h2 = u2; d0 = e0; d1 = e1; d2 = e2;
    }
    gact(h0, d0, T1[0]); gact(h1, d1, T1[1]); gact(h2, d2, T1[2]);

    // ----- layer 2: 3 -> 3 -----
    {
      float u0 = fmaf(W2[0], h0, fmaf(W2[1], h1, fmaf(W2[2], h2, B2[0])));
      float u1 = fmaf(W2[3], h0, fmaf(W2[4], h1, fmaf(W2[5], h2, B2[1])));
      float u2 = fmaf(W2[6], h0, fmaf(W2[7], h1, fmaf(W2[8], h2, B2[2])));
      float e0 = fmaf(W2[0], d0, fmaf(W2[1], d1, W2[2] * d2));
      float e1 = fmaf(W2[3], d0, fmaf(W2[4], d1, W2[5] * d2));
      float e2 = fmaf(W2[6], d0, fmaf(W2[7], d1, W2[8] * d2));
      h0 = u0; h1 = u1; h2 = u2; d0 = e0; d1 = e1; d2 = e2;
    }
    gact(h0, d0, T2[0]); gact(h1, d1, T2[1]); gact(h2, d2, T2[2]);

    // ----- layer 3: 3 -> 3 -----
    {
      float u0 = fmaf(W3[0], h0, fmaf(W3[1], h1, fmaf(W3[2], h2, B3[0])));
      float u1 = fmaf(W3[3], h0, fmaf(W3[4], h1, fmaf(W3[5], h2, B3[1])));
      float u2 = fmaf(W3[6], h0, fmaf(W3[7], h1, fmaf(W3[8], h2, B3[2])));
      float e0 = fmaf(W3[0], d0, fmaf(W3[1], d1, W3[2] * d2));
      float e1 = fmaf(W3[3], d0, fmaf(W3[4], d1, W3[5] * d2));
      float e2 = fmaf(W3[6], d0, fmaf(W3[7], d1, W3[8] * d2));
      h0 = u0; h1 = u1; h2 = u2; d0 = e0; d1 = e1; d2 = e2;
    }
    gact(h0, d0, T3[0]); gact(h1, d1, T3[1]); gact(h2, d2, T3[2]);

    // ----- layer 4: 3 -> 1, then sigmoid' * dL/dx -----
    const float L  = fmaf(W4[0], h0, fmaf(W4[1], h1, fmaf(W4[2], h2, B4)));
    const float dL = fmaf(W4[0], d0, fmaf(W4[1], d1, W4[2] * d2));
    const float s  = fast_rcp(1.0f + fast_exp2(-LOG2E * L));  // sigmoid(L)
    acc = fmaf(s * (1.0f - s), dL, acc);                       // pdf = s(1-s)L'
  }

  out[b * CCH + c] = acc * (1.0f / (float)NNN);
}

// ---------------------------------------------------------------------------
// Prepass: fold softplus(m_i), tanh(f_i) into a compact per-channel record.
// One thread per channel; trivial cost (256 threads, ~45 transcendentals).
// ---------------------------------------------------------------------------
__global__ void ebl_prep_params(
    const float* __restrict__ m0, const float* __restrict__ b0, const float* __restrict__ f0,
    const float* __restrict__ m1, const float* __restrict__ b1, const float* __restrict__ f1,
    const float* __restrict__ m2, const float* __restrict__ b2, const float* __restrict__ f2,
    const float* __restrict__ m3, const float* __restrict__ b3, const float* __restrict__ f3,
    const float* __restrict__ m4, const float* __restrict__ b4,
    float* __restrict__ rec) {
  const int c = blockIdx.x * blockDim.x + threadIdx.x;
  if (c >= CCH) return;
  float* r = rec + c * REC;
#pragma unroll
  for (int i = 0; i < 3; ++i) r[0 + i]  = softplus_f(m0[c * 3 + i]);
#pragma unroll
  for (int i = 0; i < 9; ++i) r[3 + i]  = softplus_f(m1[c * 9 + i]);
#pragma unroll
  for (int i = 0; i < 9; ++i) r[12 + i] = softplus_f(m2[c * 9 + i]);
#pragma unroll
  for (int i = 0; i < 9; ++i) r[21 + i] = softplus_f(m3[c * 9 + i]);
#pragma unroll
  for (int i = 0; i < 3; ++i) r[30 + i] = softplus_f(m4[c * 3 + i]);
#pragma unroll
  for (int i = 0; i < 3; ++i) r[33 + i] = b0[c * 3 + i];
#pragma unroll
  for (int i = 0; i < 3; ++i) r[36 + i] = b1[c * 3 + i];
#pragma unroll
  for (int i = 0; i < 3; ++i) r[39 + i] = b2[c * 3 + i];
#pragma unroll
  for (int i = 0; i < 3; ++i) r[42 + i] = b3[c * 3 + i];
  r[45] = b4[c];
#pragma unroll
  for (int i = 0; i < 3; ++i) r[46 + i] = fast_tanh(f0[c * 3 + i]);
#pragma unroll
  for (int i = 0; i < 3; ++i) r[49 + i] = fast_tanh(f1[c * 3 + i]);
#pragma unroll
  for (int i = 0; i < 3; ++i) r[52 + i] = fast_tanh(f2[c * 3 + i]);
#pragma unroll
  for (int i = 0; i < 3; ++i) r[55 + i] = fast_tanh(f3[c * 3 + i]);
  r[58] = 0.f; r[59] = 0.f; r[60] = 0.f; r[61] = 0.f; r[62] = 0.f; r[63] = 0.f;
}

extern "C" void kernel_launch(void* const* d_in, const int* in_sizes, int n_in,
                              void* d_out, int out_size, void* d_ws, size_t ws_size,
                              hipStream_t stream) {
  (void)in_sizes; (void)n_in; (void)out_size; (void)ws_size;
  // setup_inputs() dict order:
  // 0:inputs 1:noise 2:m0 3:b0 4:f0 5:m1 6:b1 7:f1 8:m2 9:b2 10:f2
  // 11:m3 12:b3 13:f3 14:m4 15:b4
  const float* inputs = (const float*)d_in[0];
  const float* noise  = (const float*)d_in[1];
  const float* m0 = (const float*)d_in[2];
  const float* b0 = (const float*)d_in[3];
  const float* f0 = (const float*)d_in[4];
  const float* m1 = (const float*)d_in[5];
  const float* b1 = (const float*)d_in[6];
  const float* f1 = (const float*)d_in[7];
  const float* m2 = (const float*)d_in[8];
  const float* b2 = (const float*)d_in[9];
  const float* f2 = (const float*)d_in[10];
  const float* m3 = (const float*)d_in[11];
  const float* b3 = (const float*)d_in[12];
  const float* f3 = (const float*)d_in[13];
  const float* m4 = (const float*)d_in[14];
  const float* b4 = (const float*)d_in[15];
  float* out = (float*)d_out;
  float* rec = (float*)d_ws;   // needs CCH*REC*4 = 64 KB of scratch

  ebl_prep_params<<<1, CCH, 0, stream>>>(m0, b0, f0, m1, b1, f1, m2, b2, f2,
                                         m3, b3, f3, m4, b4, rec);
  ebl_main<<<CCH * BLOCKS_PER_C, BLK, 0, stream>>>(inputs, noise, rec, out);
}